// EdgePredictor_6906307412165
// MI455X (gfx1250) — compile-verified
//
#include <hip/hip_runtime.h>
#include <hip/hip_bf16.h>
#include <math.h>
#include <stdint.h>

// ---------------------------------------------------------------------------
// Problem constants (from reference setup_inputs)
// ---------------------------------------------------------------------------
#define NNODE 4096
#define INDIM 3
#define HDIM  64
#define NEDGE 24576
#define KNN   15
#define ETOT  (NEDGE + NNODE * KNN)   // 86016

typedef float v2f __attribute__((ext_vector_type(2)));
typedef float v8f __attribute__((ext_vector_type(8)));

// ---------------------------------------------------------------------------
// gfx1250 async global->LDS copy (ASYNCcnt path), with sync fallback.
// Builtin signature (per hipcc diagnostic): (int4 as1*, int4 as3*, imm, imm).
// LDS "generic" pointers keep the LDS byte offset in the low 32 bits
// (flat aperture rules), so an integer round-trip yields a valid as(3) ptr.
// ---------------------------------------------------------------------------
#define AS1 __attribute__((address_space(1)))
#define AS3 __attribute__((address_space(3)))
typedef int i4v __attribute__((vector_size(4 * sizeof(int))));

#if defined(__has_builtin)
#if __has_builtin(__builtin_amdgcn_global_load_async_to_lds_b128) && \
    __has_builtin(__builtin_amdgcn_s_wait_asynccnt)
#define USE_ASYNC_COPY 1
#endif
#endif
#ifndef USE_ASYNC_COPY
#define USE_ASYNC_COPY 0
#endif

__device__ __forceinline__ void copy16_g2lds(const float* g, float* l) {
#if USE_ASYNC_COPY
  __builtin_amdgcn_global_load_async_to_lds_b128(
      (AS1 i4v*)(uintptr_t)g, (AS3 i4v*)(uint32_t)(uintptr_t)l, 0, 0);
#else
  *(float4*)l = *(const float4*)g;
#endif
}
__device__ __forceinline__ void copy_join() {
#if USE_ASYNC_COPY
  __builtin_amdgcn_s_wait_asynccnt(0);
#endif
}

// ---------------------------------------------------------------------------
// float atomic max via signed-int max / unsigned-int min trick (total order on
// float bit patterns). agg/m buffers are initialized to -inf (0xFF800000).
// ---------------------------------------------------------------------------
__device__ __forceinline__ void atomicMaxF32(float* addr, float v) {
  if (v >= 0.0f) atomicMax((int*)addr, __float_as_int(v));
  else           atomicMin((unsigned int*)addr, __float_as_uint(v));
}

// ---------------------------------------------------------------------------
// fill kernel (bit pattern) -- used for zeroing S / A and -inf init
// ---------------------------------------------------------------------------
__global__ void fill_u32(unsigned int* __restrict__ p, unsigned int pat, int n) {
  int i = blockIdx.x * blockDim.x + threadIdx.x;
  if (i < n) p[i] = pat;
}

// ---------------------------------------------------------------------------
// kNN: one thread per node, whole point cloud (48 KB) resident in LDS.
// Exact top-15 by insertion into fully-unrolled register arrays.
// ---------------------------------------------------------------------------
__global__ void knn_kernel(const float* __restrict__ x, int* __restrict__ knn_col) {
  __shared__ float xs[NNODE * 3];
  for (int t = threadIdx.x; t < NNODE * 3; t += blockDim.x) xs[t] = x[t];
  __syncthreads();

  int i = blockIdx.x * blockDim.x + threadIdx.x;   // 32 blocks x 128 threads
  float xi0 = xs[3 * i + 0], xi1 = xs[3 * i + 1], xi2 = xs[3 * i + 2];

  float bd[KNN];
  int   bi[KNN];
#pragma unroll
  for (int t = 0; t < KNN; ++t) { bd[t] = INFINITY; bi[t] = 0; }

  for (int j = 0; j < NNODE; ++j) {
    float d0 = xs[3 * j + 0] - xi0;
    float d1 = xs[3 * j + 1] - xi1;
    float d2 = xs[3 * j + 2] - xi2;
    float d  = d0 * d0 + d1 * d1 + d2 * d2;
    if (j == i) continue;                 // loop=False: exclude self
    if (d < bd[KNN - 1]) {
      int p = 0;
#pragma unroll
      for (int t = 0; t < KNN - 1; ++t) if (d >= bd[t]) p = t + 1;
#pragma unroll
      for (int t = KNN - 1; t >= 1; --t)
        if (t > p) { bd[t] = bd[t - 1]; bi[t] = bi[t - 1]; }
#pragma unroll
      for (int t = 0; t < KNN; ++t)
        if (t == p) { bd[t] = d; bi[t] = j; }
    }
  }
#pragma unroll
  for (int t = 0; t < KNN; ++t) knn_col[i * KNN + t] = bi[t];
}

// ---------------------------------------------------------------------------
// DevConv message + segment_max: one thread per (edge, channel).
// msg = Wt[h,:] . (x[c] - x[r]); agg[r,h] = max(agg[r,h], msg)
// ---------------------------------------------------------------------------
__global__ void devconv_kernel(const float* __restrict__ x,
                               const int* __restrict__ edges,
                               const int* __restrict__ knn_col,
                               const float* __restrict__ Wt,
                               float* __restrict__ agg) {
  __shared__ float wt[HDIM * 3];
  for (int t = threadIdx.x; t < HDIM * 3; t += blockDim.x) wt[t] = Wt[t];
  __syncthreads();

  long id = (long)blockIdx.x * blockDim.x + threadIdx.x;
  if (id >= (long)ETOT * HDIM) return;
  int e = (int)(id >> 6);
  int h = (int)(id & 63);

  int r, c;
  if (e < NEDGE) { r = edges[e]; c = edges[NEDGE + e]; }
  else           { int t = e - NEDGE; r = t / KNN; c = knn_col[t]; }

  float d0 = x[3 * c + 0] - x[3 * r + 0];
  float d1 = x[3 * c + 1] - x[3 * r + 1];
  float d2 = x[3 * c + 2] - x[3 * r + 2];
  float m  = wt[h * 3 + 0] * d0 + wt[h * 3 + 1] * d1 + wt[h * 3 + 2] * d2;
  atomicMaxF32(&agg[r * HDIM + h], m);
}

// ---------------------------------------------------------------------------
// Small dense Y[N,64] = X[N,64] @ W[64,64]^T (torch Linear convention).
// Optionally replaces non-finite X entries with 0 (post-segment-max fixup).
// 64 rows per 256-thread block; X tile and W cached in LDS. ~33 MFLOP total.
// ---------------------------------------------------------------------------
__global__ void ngemm64_kernel(const float* __restrict__ X,
                               const float* __restrict__ W,
                               float* __restrict__ Y, int maskNonFinite) {
  __shared__ float Ws[HDIM * HDIM];
  __shared__ float Xs[64 * HDIM];
  int r0 = blockIdx.x * 64;
  for (int t = threadIdx.x; t < HDIM * HDIM; t += blockDim.x) Ws[t] = W[t];
  for (int t = threadIdx.x; t < 64 * HDIM; t += blockDim.x) {
    float v = X[r0 * HDIM + t];
    if (maskNonFinite) {
      unsigned u = __float_as_uint(v);
      if ((u & 0x7F800000u) == 0x7F800000u) v = 0.0f;   // Inf / NaN -> 0
    }
    Xs[t] = v;
  }
  __syncthreads();

  int r  = threadIdx.x >> 2;            // 0..63
  int c0 = (threadIdx.x & 3) * 16;      // 4 threads per row, 16 cols each
  for (int c = c0; c < c0 + 16; ++c) {
    float acc = 0.0f;
#pragma unroll
    for (int k = 0; k < HDIM; ++k) acc += Xs[r * HDIM + k] * Ws[c * HDIM + k];
    Y[(r0 + r) * HDIM + c] = acc;
  }
}

// ---------------------------------------------------------------------------
// Attention over original edges
// ---------------------------------------------------------------------------
__global__ void att_score_kernel(const float* __restrict__ q,
                                 const float* __restrict__ kf,
                                 const int* __restrict__ edges,
                                 float* __restrict__ att,
                                 float* __restrict__ mrow) {
  int e = blockIdx.x * blockDim.x + threadIdx.x;
  if (e >= NEDGE) return;
  int r = edges[e], c = edges[NEDGE + e];
  const float4* qp = (const float4*)(q  + r * HDIM);
  const float4* kp = (const float4*)(kf + c * HDIM);
  float acc = 0.0f;
#pragma unroll
  for (int t = 0; t < HDIM / 4; ++t) {
    float4 a = qp[t], b = kp[t];
    acc += a.x * b.x + a.y * b.y + a.z * b.z + a.w * b.w;
  }
  att[e] = acc;
  atomicMaxF32(&mrow[r], acc);
}

__global__ void att_exp_kernel(const int* __restrict__ edges,
                               float* __restrict__ att,
                               const float* __restrict__ mrow,
                               float* __restrict__ den) {
  int e = blockIdx.x * blockDim.x + threadIdx.x;
  if (e >= NEDGE) return;
  int r = edges[e];
  float ex = expf(att[e] - mrow[r]);
  att[e] = ex;
  atomicAdd(&den[r], ex);
}

// S[r,c] += attn ; A[r,c] += 1  (duplicate edges coalesce by summation)
__global__ void scatter_SA_kernel(const int* __restrict__ edges,
                                  const float* __restrict__ att,
                                  const float* __restrict__ den,
                                  float* __restrict__ S,
                                  float* __restrict__ A) {
  int e = blockIdx.x * blockDim.x + threadIdx.x;
  if (e >= NEDGE) return;
  int r = edges[e], c = edges[NEDGE + e];
  atomicAdd(&S[(long)r * NNODE + c], att[e] / den[r]);
  atomicAdd(&A[(long)r * NNODE + c], 1.0f);
}

// ---------------------------------------------------------------------------
// Dense fp32 GEMM with V_WMMA_F32_16X16X4_F32 (the only f32 WMMA shape).
//   TRANSB = false : C = Aop @ Bop         (both row-major NxN)
//   TRANSB = true  : C = Aop @ Bop^T       (C[i,j] = sum_k Aop[i,k]*Bop[j,k])
//
// Block: 256 threads (8 x wave32), 64x64 C tile, K-tile 32.
// Each wave owns two 16x16 subtiles.
//
// Both variants stage B in a unified LDS layout Bs[n][k] (row stride 36
// floats) so every WMMA fragment -- A pair and both B pairs -- is a single
// contiguous 8-byte LDS read straight into an even-aligned VGPR pair
// (ds_load_b64, no repacking movs in the hot loop).
//   * A tile and TRANSB B tile: contiguous 16B global chunks -> contiguous
//     16B LDS chunks => async DMA (GLOBAL_LOAD_ASYNC_TO_LDS_B128, ASYNCcnt).
//   * TRANSB=false B tile is transposed during load (scalar ds stores, once
//     per K-tile, outside the WMMA loop).
//
// Fragment layouts per CDNA5 ISA 7.12.2 (16x4 f32 A; mirrored for 4x16 B):
//   A: lane<16 -> row=lane,   v2f = {A[m][k+0], A[m][k+1]}
//      lane>=16-> row=lane-16, v2f = {A[m][k+2], A[m][k+3]}
//   C/D: VGPR v -> (row = v + (lane<16 ? 0 : 8), col = lane%16)
// ---------------------------------------------------------------------------
template <bool TRANSB>
__global__ void wmma_gemm_kernel(const float* __restrict__ Aop,
                                 const float* __restrict__ Bop,
                                 float* __restrict__ Cop) {
  constexpr int BM = 64, BN = 64, BK = 32;
  constexpr int LD = BK + 4;                    // 36 floats = 144 B (16B mult)
  __shared__ __align__(16) float As[BM * LD];   // 9216 B, [m][k]
  __shared__ __align__(16) float Bs[BN * LD];   // 9216 B, [n][k]

  const int Kdim = NNODE, Ncol = NNODE;
  const int tid  = threadIdx.x;
  const int lane = tid & 31;
  const int wave = tid >> 5;
  const int lm   = lane & 15;
  const int khi  = (lane >> 4) << 1;            // 0 or 2
  const int mo   = (wave & 3) * 16;             // wave's subtile row
  const int no0  = (wave >> 2) * 32;            // wave's subtile cols
  const int no1  = no0 + 16;
  const int m0g  = blockIdx.y * BM;
  const int n0g  = blockIdx.x * BN;

  v8f acc0 = {0.f, 0.f, 0.f, 0.f, 0.f, 0.f, 0.f, 0.f};
  v8f acc1 = {0.f, 0.f, 0.f, 0.f, 0.f, 0.f, 0.f, 0.f};

  for (int k0 = 0; k0 < Kdim; k0 += BK) {
    // ---- A tile 64x32: 2 x 16B per thread, async DMA to LDS ----
#pragma unroll
    for (int v = 0; v < 2; ++v) {
      int fid = tid * 2 + v;                    // 0..511
      int m   = fid >> 3;
      int kk  = (fid & 7) << 2;
      copy16_g2lds(Aop + (long)(m0g + m) * Kdim + k0 + kk, &As[m * LD + kk]);
    }
    // ---- B tile -> Bs[n][k] ----
    if (TRANSB) {
#pragma unroll
      for (int v = 0; v < 2; ++v) {             // layout-compatible: async DMA
        int fid = tid * 2 + v;
        int n   = fid >> 3;
        int kk  = (fid & 7) << 2;
        copy16_g2lds(Bop + (long)(n0g + n) * Kdim + k0 + kk, &Bs[n * LD + kk]);
      }
    } else {
#pragma unroll
      for (int v = 0; v < 2; ++v) {             // transpose during load
        int fid = tid * 2 + v;
        int kk  = fid >> 4;                     // 0..31
        int n   = (fid & 15) << 2;              // 0..60
        float4 t = *(const float4*)(Bop + (long)(k0 + kk) * Ncol + n0g + n);
        Bs[(n + 0) * LD + kk] = t.x;
        Bs[(n + 1) * LD + kk] = t.y;
        Bs[(n + 2) * LD + kk] = t.z;
        Bs[(n + 3) * LD + kk] = t.w;
      }
    }
    copy_join();                                // s_wait_asynccnt 0
    __syncthreads();

    // ---- prefetch next K tile into L2 (global_prefetch_b8) ----
    if (k0 + BK < Kdim) {
      int fid = tid * 2;
      int m   = fid >> 3;
      int kk  = (fid & 7) << 2;
      __builtin_prefetch(Aop + (long)(m0g + m) * Kdim + k0 + BK + kk, 0, 3);
      if (TRANSB) {
        __builtin_prefetch(Bop + (long)(n0g + m) * Kdim + k0 + BK + kk, 0, 3);
      } else {
        int kb = fid >> 4, nb = (fid & 15) << 2;
        __builtin_prefetch(Bop + (long)(k0 + BK + kb) * Ncol + n0g + nb, 0, 3);
      }
    }

    // ---- 8 k-steps x 2 WMMAs per wave; all frags are ds_load_b64 ----
#pragma unroll
    for (int kk = 0; kk < BK; kk += 4) {
      int ka = kk + khi;                        // even => 8B-aligned
      v2f a  = *(const v2f*)&As[(mo  + lm) * LD + ka];
      v2f b0 = *(const v2f*)&Bs[(no0 + lm) * LD + ka];
      v2f b1 = *(const v2f*)&Bs[(no1 + lm) * LD + ka];
      acc0 = __builtin_amdgcn_wmma_f32_16x16x4_f32(false, a, false, b0,
                                                   (short)0, acc0, false, false);
      acc1 = __builtin_amdgcn_wmma_f32_16x16x4_f32(false, a, false, b1,
                                                   (short)0, acc1, false, false);
    }
    __syncthreads();
  }

  // ---- epilogue: C/D layout -> global ----
  int mrow = m0g + mo + ((lane >> 4) << 3);
#pragma unroll
  for (int v = 0; v < 8; ++v) {
    Cop[(long)(mrow + v) * Ncol + n0g + no0 + lm] = acc0[v];
    Cop[(long)(mrow + v) * Ncol + n0g + no1 + lm] = acc1[v];
  }
}

// ---------------------------------------------------------------------------
// Host-side orchestration.
//
// Workspace layout (needs ~134 MB):
//   [0,   64MB)   S   dense 4096x4096 f32
//   [64,  128MB)  T   dense 4096x4096 f32 (T = S @ A)
//   [128MB, ...)  aux: knn_col / agg / feat / q / k / att / m / den
// A (edge-count matrix) is built directly in d_out, then overwritten by the
// final GEMM (which reads only T and S) -- saves 64 MB and a d2d copy.
// ---------------------------------------------------------------------------
extern "C" void kernel_launch(void* const* d_in, const int* in_sizes, int n_in,
                              void* d_out, int out_size, void* d_ws, size_t ws_size,
                              hipStream_t stream) {
  const float* x     = (const float*)d_in[0];
  const int*   edges = (const int*)d_in[1];
  const float* Wt    = (const float*)d_in[2];
  const float* Wp    = (const float*)d_in[3];
  const float* Wq    = (const float*)d_in[4];
  const float* Wk    = (const float*)d_in[5];
  float* Adense = (float*)d_out;          // A first, final output later

  char*  ws      = (char*)d_ws;
  float* S       = (float*)(ws);
  float* T       = (float*)(ws + ((size_t)64 << 20));
  char*  aux     = ws + ((size_t)128 << 20);
  int*   knn_col = (int*)(aux);                         // 240 KB
  float* agg     = (float*)(aux + (1 << 20));           // 1 MB
  float* feat    = (float*)(aux + (2 << 20));
  float* qb      = (float*)(aux + (3 << 20));
  float* kb      = (float*)(aux + (4 << 20));
  float* att     = (float*)(aux + (5 << 20));           // 96 KB
  float* mrow    = (float*)(aux + (5 << 20) + (256 << 10));
  float* den     = (float*)(aux + (5 << 20) + (384 << 10));

  const int NN = NNODE * NNODE;

  // init (harness poisons d_out/d_ws; every launch must re-init)
  fill_u32<<<(NN + 255) / 256, 256, 0, stream>>>((unsigned*)S, 0u, NN);
  fill_u32<<<(NN + 255) / 256, 256, 0, stream>>>((unsigned*)Adense, 0u, NN);
  fill_u32<<<(NNODE * HDIM + 255) / 256, 256, 0, stream>>>(
      (unsigned*)agg, 0xFF800000u, NNODE * HDIM);                 // -inf
  fill_u32<<<(NNODE + 255) / 256, 256, 0, stream>>>((unsigned*)mrow,
                                                    0xFF800000u, NNODE);
  fill_u32<<<(NNODE + 255) / 256, 256, 0, stream>>>((unsigned*)den, 0u, NNODE);

  // 1) kNN graph
  knn_kernel<<<NNODE / 128, 128, 0, stream>>>(x, knn_col);

  // 2) DevConv edge messages + segment max over union edge set
  long dv = (long)ETOT * HDIM;
  devconv_kernel<<<(int)((dv + 255) / 256), 256, 0, stream>>>(x, edges, knn_col,
                                                              Wt, agg);

  // 3) feat = mask(agg) @ Wp^T ; q = feat @ Wq^T ; k = feat @ Wk^T
  ngemm64_kernel<<<NNODE / 64, 256, 0, stream>>>(agg, Wp, feat, 1);
  ngemm64_kernel<<<NNODE / 64, 256, 0, stream>>>(feat, Wq, qb, 0);
  ngemm64_kernel<<<NNODE / 64, 256, 0, stream>>>(feat, Wk, kb, 0);

  // 4) scatter-softmax attention over original edges
  att_score_kernel<<<(NEDGE + 255) / 256, 256, 0, stream>>>(qb, kb, edges, att,
                                                            mrow);
  att_exp_kernel<<<(NEDGE + 255) / 256, 256, 0, stream>>>(edges, att, mrow, den);
  scatter_SA_kernel<<<(NEDGE + 255) / 256, 256, 0, stream>>>(edges, att, den, S,
                                                             Adense);

  // 5) out = (S @ A) @ S^T  -- two 4096^3 fp32 WMMA GEMMs (dominant cost)
  dim3 g(NNODE / 64, NNODE / 64);
  wmma_gemm_kernel<false><<<g, 256, 0, stream>>>(S, Adense, T);        // T = S@A
  wmma_gemm_kernel<true ><<<g, 256, 0, stream>>>(T, S, (float*)d_out); // T@S^T

  (void)in_sizes; (void)n_in; (void)out_size; (void)ws_size;
}